// SAGE_VE_54228257080133
// MI455X (gfx1250) — compile-verified
//
#include <hip/hip_runtime.h>

#define N_NODES 100000
#define N_EDGES 5000000
#define IN_DIM 16
#define EMB_DIM 8

typedef __attribute__((ext_vector_type(2))) float v2f;
typedef __attribute__((ext_vector_type(8))) float v8f;

// ---------------- zero fill ----------------
__global__ void sage_zero_f32(float* __restrict__ p, long long n) {
    long long i = (long long)blockIdx.x * blockDim.x + threadIdx.x;
    if (i < n) p[i] = 0.0f;
}

// ---------------- layer-1 edge scatter: acc[dst] += w * x[src], cnt[dst] += 1 ----------------
__global__ void sage_scatter16(const float* __restrict__ x,
                               const long long* __restrict__ src,
                               const long long* __restrict__ dst,
                               const float* __restrict__ ew,
                               float* __restrict__ acc,
                               float* __restrict__ cnt) {
    int e = blockIdx.x * blockDim.x + threadIdx.x;
    if (e >= N_EDGES) return;
    long long s = src[e];
    long long d = dst[e];
    float w = ew[e];
    const float4* xr = (const float4*)(x + s * IN_DIM);  // one 64B line, L2-resident
    float* a = acc + d * IN_DIM;
#pragma unroll
    for (int q = 0; q < 4; ++q) {
        float4 v = xr[q];
        atomicAdd(a + 4 * q + 0, v.x * w);
        atomicAdd(a + 4 * q + 1, v.y * w);
        atomicAdd(a + 4 * q + 2, v.z * w);
        atomicAdd(a + 4 * q + 3, v.w * w);
    }
    atomicAdd(cnt + d, 1.0f);
}

// ---------------- layer-2 edge scatter: acc[dst] += w * h[src] (counts reused) ----------------
__global__ void sage_scatter8(const float* __restrict__ h,
                              const long long* __restrict__ src,
                              const long long* __restrict__ dst,
                              const float* __restrict__ ew,
                              float* __restrict__ acc) {
    int e = blockIdx.x * blockDim.x + threadIdx.x;
    if (e >= N_EDGES) return;
    long long s = src[e];
    long long d = dst[e];
    float w = ew[e];
    const float4* hr = (const float4*)(h + s * EMB_DIM);
    float* a = acc + d * EMB_DIM;
#pragma unroll
    for (int q = 0; q < 2; ++q) {
        float4 v = hr[q];
        atomicAdd(a + 4 * q + 0, v.x * w);
        atomicAdd(a + 4 * q + 1, v.y * w);
        atomicAdd(a + 4 * q + 2, v.z * w);
        atomicAdd(a + 4 * q + 3, v.w * w);
    }
}

// ---------------- layer-1 node update via fp32 WMMA: h = relu(mean_aggr@Wl + b + x@Wr) ----------------
__global__ void __launch_bounds__(256)
sage_node1(const float* __restrict__ x,
           const float* __restrict__ acc,
           const float* __restrict__ cnt,
           const float* __restrict__ Wl,   // [16][8] row-major
           const float* __restrict__ Wr,   // [16][8]
           const float* __restrict__ bias, // [8]
           float* __restrict__ h) {
    const int ntiles = N_NODES / 16;                       // 6250, exact
    int wave = (blockIdx.x * blockDim.x + threadIdx.x) >> 5;
    if (wave >= ntiles) return;                            // wave-uniform: EXEC stays all-1s
    int lane = threadIdx.x & 31;
    int m = lane & 15;                                     // A row / B,C column
    int halfk = lane >> 4;                                 // which K pair this lane supplies
    int mc = m & 7;                                        // clamped column (in-bounds load)
    float msk = (m < 8) ? 1.0f : 0.0f;                     // zero-pad B cols 8..15 via VALU
    int node = wave * 16 + m;

    float inv = 1.0f / fmaxf(cnt[node], 1.0f);
    const float* arow = acc + (long long)node * IN_DIM;
    const float* xrow = x + (long long)node * IN_DIM;

    // Build all A/B fragments first (unconditional loads -> no exec divergence),
    // then run the 8-deep WMMA chain back-to-back.
    v2f aL[4], aR[4], bL[4], bR[4];
#pragma unroll
    for (int s = 0; s < 4; ++s) {
        int k = 4 * s + 2 * halfk;
        aL[s].x = arow[k] * inv;     aL[s].y = arow[k + 1] * inv;
        aR[s].x = xrow[k];           aR[s].y = xrow[k + 1];
        bL[s].x = Wl[k * EMB_DIM + mc] * msk;
        bL[s].y = Wl[(k + 1) * EMB_DIM + mc] * msk;
        bR[s].x = Wr[k * EMB_DIM + mc] * msk;
        bR[s].y = Wr[(k + 1) * EMB_DIM + mc] * msk;
    }

    v8f c = {};
#pragma unroll
    for (int s = 0; s < 4; ++s)                            // aggr @ Wl, K = 16 in 4 steps
        c = __builtin_amdgcn_wmma_f32_16x16x4_f32(false, aL[s], false, bL[s], (short)0, c, false, false);
#pragma unroll
    for (int s = 0; s < 4; ++s)                            // + x @ Wr into same accumulator
        c = __builtin_amdgcn_wmma_f32_16x16x4_f32(false, aR[s], false, bR[s], (short)0, c, false, false);

    float bv = bias[mc];                                   // unconditional; used only if m<8
    if (m < 8) {                                           // predicate only the stores
#pragma unroll
        for (int j = 0; j < 8; ++j) {
            int row = wave * 16 + j + 8 * halfk;           // C VGPR j: M = j + 8*halfk
            float v = c[j] + bv;
            h[(long long)row * EMB_DIM + m] = fmaxf(v, 0.0f);
        }
    }
}

// ---------------- layer-2 node update via fp32 WMMA + log_softmax ----------------
__global__ void __launch_bounds__(256)
sage_node2(const float* __restrict__ hx,
           const float* __restrict__ acc,
           const float* __restrict__ cnt,
           const float* __restrict__ Wl,   // [8][8]
           const float* __restrict__ Wr,   // [8][8]
           const float* __restrict__ bias, // [8]
           float* __restrict__ out) {
    const int ntiles = N_NODES / 16;
    int wave = (blockIdx.x * blockDim.x + threadIdx.x) >> 5;
    if (wave >= ntiles) return;                            // wave-uniform
    int lane = threadIdx.x & 31;
    int m = lane & 15;
    int halfk = lane >> 4;
    int mc = m & 7;
    float msk = (m < 8) ? 1.0f : 0.0f;
    int node = wave * 16 + m;

    float inv = 1.0f / fmaxf(cnt[node], 1.0f);
    const float* arow = acc + (long long)node * EMB_DIM;
    const float* hrow = hx + (long long)node * EMB_DIM;

    v2f aL[2], aR[2], bL[2], bR[2];
#pragma unroll
    for (int s = 0; s < 2; ++s) {
        int k = 4 * s + 2 * halfk;
        aL[s].x = arow[k] * inv;     aL[s].y = arow[k + 1] * inv;
        aR[s].x = hrow[k];           aR[s].y = hrow[k + 1];
        bL[s].x = Wl[k * EMB_DIM + mc] * msk;
        bL[s].y = Wl[(k + 1) * EMB_DIM + mc] * msk;
        bR[s].x = Wr[k * EMB_DIM + mc] * msk;
        bR[s].y = Wr[(k + 1) * EMB_DIM + mc] * msk;
    }

    v8f c = {};
#pragma unroll
    for (int s = 0; s < 2; ++s)                            // aggr @ W3l, K = 8 in 2 steps
        c = __builtin_amdgcn_wmma_f32_16x16x4_f32(false, aL[s], false, bL[s], (short)0, c, false, false);
#pragma unroll
    for (int s = 0; s < 2; ++s)                            // + h @ W3r
        c = __builtin_amdgcn_wmma_f32_16x16x4_f32(false, aR[s], false, bR[s], (short)0, c, false, false);

    // Row-wise log_softmax: the 8 logits of one row live in lanes {g*16 + 0..7}
    // at the same C VGPR index j -> width-8 shuffle butterfly. Shuffle groups of 8
    // are uniformly active/inactive under the (m < 8) predicate.
    float bv = bias[mc];
    if (m < 8) {
#pragma unroll
        for (int j = 0; j < 8; ++j) {
            int row = wave * 16 + j + 8 * halfk;
            float v = c[j] + bv;
            float mx = v;
#pragma unroll
            for (int mask = 1; mask < 8; mask <<= 1)
                mx = fmaxf(mx, __shfl_xor(mx, mask, 8));
            float ex = __expf(v - mx);
            float sm = ex;
#pragma unroll
            for (int mask = 1; mask < 8; mask <<= 1)
                sm += __shfl_xor(sm, mask, 8);
            out[(long long)row * EMB_DIM + m] = (v - mx) - __logf(sm);
        }
    }
}

extern "C" void kernel_launch(void* const* d_in, const int* in_sizes, int n_in,
                              void* d_out, int out_size, void* d_ws, size_t ws_size,
                              hipStream_t stream) {
    const float* x        = (const float*)d_in[0];
    const long long* ei   = (const long long*)d_in[1];   // int64 [2, E]
    const float* ew       = (const float*)d_in[2];
    const float* W1l      = (const float*)d_in[3];
    const float* b1       = (const float*)d_in[4];
    const float* W1r      = (const float*)d_in[5];
    const float* W3l      = (const float*)d_in[6];
    const float* b3       = (const float*)d_in[7];
    const float* W3r      = (const float*)d_in[8];
    float* out            = (float*)d_out;

    const long long* srcp = ei;
    const long long* dstp = ei + N_EDGES;

    // workspace: [cnt: N][acc: N*16 (reused as N*8)][h: N*8]  = 10 MB, all L2-resident
    float* ws  = (float*)d_ws;
    float* cnt = ws;
    float* acc = ws + N_NODES;
    float* h   = ws + N_NODES + (long long)N_NODES * IN_DIM;

    // --- layer 1 ---
    {
        long long nz = (long long)N_NODES * (1 + IN_DIM);  // cnt + acc contiguous
        sage_zero_f32<<<(unsigned)((nz + 255) / 256), 256, 0, stream>>>(ws, nz);
    }
    sage_scatter16<<<(N_EDGES + 255) / 256, 256, 0, stream>>>(x, srcp, dstp, ew, acc, cnt);

    int nwaves = N_NODES / 16;                 // one wave per 16-node tile
    int nthreads = nwaves * 32;
    sage_node1<<<(nthreads + 255) / 256, 256, 0, stream>>>(x, acc, cnt, W1l, W1r, b1, h);

    // --- layer 2 (counts reused; acc re-zeroed at N*8) ---
    {
        long long nz = (long long)N_NODES * EMB_DIM;
        sage_zero_f32<<<(unsigned)((nz + 255) / 256), 256, 0, stream>>>(acc, nz);
    }
    sage_scatter8<<<(N_EDGES + 255) / 256, 256, 0, stream>>>(h, srcp, dstp, ew, acc);
    sage_node2<<<(nthreads + 255) / 256, 256, 0, stream>>>(h, acc, cnt, W3l, W3r, b3, out);
}